// FCOSDecoder_19267223290025
// MI455X (gfx1250) — compile-verified
//
#include <hip/hip_runtime.h>
#include <hip/hip_bf16.h>
#include <stdint.h>

// ---------------- problem constants (match reference) ----------------
#define BATCHN      16
#define NUM_CLASSES 80
#define IMAGE_WC    1024
#define IMAGE_HC    1024
#define TOP_NC      1000
#define MIN_SCOREC  0.01f
#define NMS_THRC    0.6f
#define MAX_DETC    100

#define NANCH 21824      // anchors per image (16384+4096+1024+256+64)
#define NCAND 4096       // padded candidate count for bitonic sort
#define NVAL  3320       // real candidates (1000*3 + 256 + 64)
#define NROW  3328       // NVAL padded to multiple of 32
#define NW    104        // 32-bit mask words per row (3328/32)

typedef unsigned long long u64;

// ---------------- CDNA5 async global->LDS helpers ----------------
__device__ __forceinline__ void async_b128(unsigned lds_off, const void* gaddr) {
  asm volatile("global_load_async_to_lds_b128 %0, %1, off"
               :: "v"(lds_off), "v"(gaddr) : "memory");
}
__device__ __forceinline__ void wait_async_all() {
  asm volatile("s_wait_asynccnt 0" ::: "memory");
}

// ---------------- helpers ----------------
__device__ __forceinline__ unsigned fmap(float f) {          // order-preserving float->u32
  unsigned u = __float_as_uint(f);
  return (u & 0x80000000u) ? ~u : (u | 0x80000000u);
}
__device__ __forceinline__ float funmap(unsigned m) {
  unsigned u = (m & 0x80000000u) ? (m & 0x7FFFFFFFu) : ~m;
  return __uint_as_float(u);
}
__device__ __forceinline__ u64 pack_sa(float s, unsigned anchor) {
  // score in [0,1) -> positive float bits are monotone; ~anchor gives idx-ascending tie-break
  return ((u64)__float_as_uint(s) << 32) | (u64)(0xFFFFFFFFu - anchor);
}

// in-LDS bitonic sort of region [base, base+m); desc=true => descending
__device__ void bitonic_region(u64* sm, int base, int m, bool desc, int tid, int nt) {
  for (int k = 2; k <= m; k <<= 1) {
    for (int j = k >> 1; j > 0; j >>= 1) {
      __syncthreads();
      for (int t = tid; t < (m >> 1); t += nt) {
        int i = ((t & ~(j - 1)) << 1) | (t & (j - 1));
        int l = i + j;
        u64 a = sm[base + i], c = sm[base + l];
        bool blockDesc = (((i & k) == 0) == desc);
        bool sw = blockDesc ? (a < c) : (a > c);
        if (sw) { sm[base + i] = c; sm[base + l] = a; }
      }
    }
  }
  __syncthreads();
}

// merge a bitonic sequence of length m (desc then asc halves) into descending order
__device__ void bitonic_merge_desc(u64* sm, int m, int tid, int nt) {
  for (int j = m >> 1; j > 0; j >>= 1) {
    __syncthreads();
    for (int t = tid; t < (m >> 1); t += nt) {
      int i = ((t & ~(j - 1)) << 1) | (t & (j - 1));
      int l = i + j;
      u64 a = sm[i], c = sm[l];
      if (a < c) { sm[i] = c; sm[l] = a; }
    }
  }
  __syncthreads();
}

// ---------------- kernel 1: per-anchor decode ----------------
__global__ __launch_bounds__(256) void decode_kernel(
    const float* __restrict__ cls, const float* __restrict__ reg,
    const float* __restrict__ ctr, int H, int W, int stride, int abase,
    float* __restrict__ score, float* __restrict__ kls, float* __restrict__ box)
{
  int idx = blockIdx.x * blockDim.x + threadIdx.x;
  int total = BATCHN * H * W;
  if (idx >= total) return;
  int hw = idx % (H * W);
  int b  = idx / (H * W);
  int h = hw / W, w = hw % W;

  const float4* cp = (const float4*)(cls + (size_t)idx * NUM_CLASSES);
  float mv = -3.4e38f; int mi = 0;
#pragma unroll
  for (int q = 0; q < NUM_CLASSES / 4; q++) {   // argmax of raw logits == argmax of sigmoid
    float4 v = cp[q];
    if (v.x > mv) { mv = v.x; mi = 4 * q + 0; }
    if (v.y > mv) { mv = v.y; mi = 4 * q + 1; }
    if (v.z > mv) { mv = v.z; mi = 4 * q + 2; }
    if (v.w > mv) { mv = v.w; mi = 4 * q + 3; }
  }
  float c = ctr[idx];
  float s = sqrtf((1.f / (1.f + expf(-mv))) * (1.f / (1.f + expf(-c))));

  float4 rg = ((const float4*)reg)[idx];
  float px = (w + 0.5f) * (float)stride, py = (h + 0.5f) * (float)stride;
  float x1f = px - expf(rg.x), y1f = py - expf(rg.y);
  float x2f = px + expf(rg.z), y2f = py + expf(rg.w);
  int x1 = (int)x1f, y1 = (int)y1f, x2 = (int)x2f, y2 = (int)y2f; // trunc like astype(int32)
  x1 = max(x1, 0); y1 = max(y1, 0);
  x2 = min(x2, IMAGE_WC - 1); y2 = min(y2, IMAGE_HC - 1);

  size_t o = (size_t)b * NANCH + abase + hw;
  score[o] = s;
  kls[o]   = (float)mi;
  ((float4*)box)[o] = make_float4((float)x1, (float)y1, (float)x2, (float)y2);
}

// ---------------- kernel 2: exact per-level top-1000 (champion merge) ----------------
__global__ __launch_bounds__(512) void topk_kernel(
    const float* __restrict__ score, int L, int abase, int slotbase,
    float* __restrict__ cscore, unsigned* __restrict__ cpay)
{
  __shared__ u64 sm[2048];
  int b = blockIdx.x, tid = threadIdx.x, nt = blockDim.x;
  const float* s = score + (size_t)b * NANCH + abase;

  for (int t = tid; t < 1024; t += nt) sm[t] = pack_sa(s[t], (unsigned)(abase + t));
  bitonic_region(sm, 0, 1024, true, tid, nt);            // champion, descending
  int nch = L >> 10;
  for (int c2 = 1; c2 < nch; c2++) {
    for (int t = tid; t < 1024; t += nt)
      sm[1024 + t] = pack_sa(s[c2 * 1024 + t], (unsigned)(abase + c2 * 1024 + t));
    bitonic_region(sm, 1024, 1024, false, tid, nt);      // challenger ascending -> bitonic 2048
    bitonic_merge_desc(sm, 2048, tid, nt);               // keep top-1024 sorted desc
  }
  for (int t = tid; t < TOP_NC; t += nt) {
    u64 v = sm[t];
    cscore[(size_t)b * NCAND + slotbase + t] = __uint_as_float((unsigned)(v >> 32));
    cpay  [(size_t)b * NCAND + slotbase + t] = 0xFFFFFFFFu - (unsigned)(v & 0xFFFFFFFFu);
  }
}

// ---------------- kernel 3: raw copy of levels 3/4 into candidate slots ----------------
__global__ void copy_tail_kernel(const float* __restrict__ score,
                                 float* __restrict__ cscore, unsigned* __restrict__ cpay)
{
  int t = blockIdx.x * blockDim.x + threadIdx.x;
  if (t >= BATCHN * 320) return;
  int b = t / 320, i = t % 320;
  int anchor = (i < 256) ? (21504 + i) : (21760 + (i - 256));
  cscore[(size_t)b * NCAND + 3000 + i] = score[(size_t)b * NANCH + anchor];
  cpay  [(size_t)b * NCAND + 3000 + i] = (unsigned)anchor;
}

// ---------------- kernel 4: per-image global sort + gather sorted box/class ----------------
__global__ __launch_bounds__(512) void sortgather_kernel(
    const float* __restrict__ cscore, const unsigned* __restrict__ cpay,
    const float* __restrict__ box, const float* __restrict__ kls,
    float* __restrict__ skey, float* __restrict__ sbox, float* __restrict__ scls)
{
  __shared__ u64 sm[NCAND];
  int b = blockIdx.x, tid = threadIdx.x, nt = blockDim.x;
  for (int t = tid; t < NCAND; t += nt) {
    float kf;
    if (t < NVAL) {
      float raw = cscore[(size_t)b * NCAND + t];
      kf = (raw > MIN_SCOREC) ? raw : -1.0f;   // invalid -> after all valid (stable by slot)
    } else kf = -2.0f;                          // padding -> last
    sm[t] = ((u64)fmap(kf) << 32) | (u64)(0xFFFFFFFFu - (unsigned)t);
  }
  bitonic_region(sm, 0, NCAND, true, tid, nt);  // (key desc, slot asc)
  for (int t = tid; t < NROW; t += nt) {
    u64 v = sm[t];
    skey[(size_t)b * NROW + t] = funmap((unsigned)(v >> 32));
    if (t < NVAL) {
      unsigned slot = 0xFFFFFFFFu - (unsigned)(v & 0xFFFFFFFFu);
      unsigned a = cpay[(size_t)b * NCAND + slot];
      size_t src = (size_t)b * NANCH + a;
      ((float4*)sbox)[(size_t)b * NROW + t] = ((const float4*)box)[src];
      scls[(size_t)b * NROW + t] = kls[src];
    } else {
      ((float4*)sbox)[(size_t)b * NROW + t] = make_float4(0.f, 0.f, 0.f, 0.f);
      scls[(size_t)b * NROW + t] = -1.f;
    }
  }
}

// ---------------- kernel 5: parallel IoU suppression bitmask ----------------
// Sorted boxes are DMA'd straight into LDS with CDNA5 async copies (no VGPR
// round-trip); each wave drains its own ASYNCcnt, then the barrier makes the
// DMA writes visible to all waves.
__global__ __launch_bounds__(256) void iou_mask_kernel(
    const float* __restrict__ sbox, unsigned* __restrict__ maskp)
{
  __shared__ __align__(16) float4 sb[NROW];     // 53 KB: all sorted boxes of this image
  int blk = blockIdx.x;
  int b = blk / NW, itile = blk % NW;           // 32 rows per block
  const char* g = (const char*)((const float4*)sbox + (size_t)b * NROW);
  unsigned ldsbase = (unsigned)(size_t)(void*)&sb[0];
#pragma unroll
  for (int t = threadIdx.x; t < NROW; t += 256) {       // 13 x b128 per lane, coalesced
    unsigned off = (unsigned)t * 16u;
    async_b128(ldsbase + off, g + off);
  }
  wait_async_all();
  __syncthreads();

  for (int t = threadIdx.x; t < 32 * NW; t += 256) {
    int r = t / NW, wjd = t % NW;
    int i = itile * 32 + r;
    float4 bi = sb[i];
    float ai = (bi.z - bi.x) * (bi.w - bi.y);
    unsigned bits = 0;
    int j0 = wjd * 32;
#pragma unroll 4
    for (int jj = 0; jj < 32; jj++) {
      int j = j0 + jj;
      if (j > i && j < NVAL) {
        float4 bj = sb[j];
        float aj = (bj.z - bj.x) * (bj.w - bj.y);
        float xx1 = fmaxf(bi.x, bj.x), yy1 = fmaxf(bi.y, bj.y);
        float xx2 = fminf(bi.z, bj.z), yy2 = fminf(bi.w, bj.w);
        float inter = fmaxf(xx2 - xx1, 0.f) * fmaxf(yy2 - yy1, 0.f);
        float iou = inter / (ai + aj - inter);  // NaN compares false, same as reference
        if (iou > NMS_THRC) bits |= 1u << jj;
      }
    }
    maskp[((size_t)b * NROW + i) * NW + wjd] = bits;
  }
}

// ---------------- kernel 6: single-wave sequential NMS scan + output ----------------
// Mask rows are double-buffer staged into LDS with CDNA5 async global->LDS copies
// (ASYNCcnt) so the serial scan never stalls on L2 latency: 26 b128s (13312 B,
// 32 mask rows) stay in flight while the previous batch is applied from LDS.
__global__ __launch_bounds__(32) void nms_out_kernel(
    const unsigned* __restrict__ maskp, const float* __restrict__ skey,
    const float* __restrict__ sbox, const float* __restrict__ scls,
    float* __restrict__ out)
{
  int b = blockIdx.x;
  int lane = threadIdx.x;
  __shared__ __align__(16) unsigned mbuf[2][32 * NW];   // 2 x 13312 B
  __shared__ unsigned keptw[NW];
  unsigned supp0 = 0, supp1 = 0, supp2 = 0, supp3 = 0;  // word w lives in suppK[w%32], K=w/32
  const char*  mbase = (const char*)(maskp + (size_t)b * NROW * NW);
  const float* keyb  = skey + (size_t)b * NROW;

  auto issue = [&](int batch, int bufsel) {
    const char* g = mbase + (size_t)batch * (32 * NW * 4);
    unsigned ldsbase = (unsigned)(size_t)(void*)&mbuf[bufsel][0];
#pragma unroll
    for (int t = 0; t < 26; t++) {              // 26 x b128/lane = 13312 B batch
      unsigned off = (unsigned)((t * 32 + lane) * 16);
      async_b128(ldsbase + off, g + off);
    }
  };
  issue(0, 0);
  issue(1, 1);

#define PROC_K(K, SUPPK)                                                      \
  for (int b32 = 0; b32 < 32; ++b32) {                                        \
    int batch = (K) * 32 + b32;                                               \
    if (batch >= NW) break;                                                   \
    if (batch < NW - 1) asm volatile("s_wait_asynccnt 26" ::: "memory");      \
    else                asm volatile("s_wait_asynccnt 0"  ::: "memory");      \
    const unsigned* buf = mbuf[batch & 1];                                    \
    float kv = keyb[batch * 32 + lane];                                       \
    unsigned validw = __builtin_amdgcn_ballot_w32(kv > MIN_SCOREC);           \
    unsigned kw = 0;                                                          \
    for (int r = 0; r < 32; ++r) {                                            \
      unsigned cw = (unsigned)__builtin_amdgcn_readlane((int)(SUPPK), b32);   \
      bool keep = ((validw >> r) & 1u) && !((cw >> r) & 1u);                  \
      if (keep) {                                                             \
        kw |= 1u << r;                                                        \
        const unsigned* row = &buf[r * NW];                                   \
        supp0 |= row[lane];                                                   \
        supp1 |= row[32 + lane];                                              \
        supp2 |= row[64 + lane];                                              \
        if (lane < 8) supp3 |= row[96 + lane];                                \
      }                                                                       \
    }                                                                         \
    if (lane == 0) keptw[batch] = kw;                                         \
    if (batch + 2 < NW) issue(batch + 2, batch & 1);                          \
  }

  PROC_K(0, supp0)
  PROC_K(1, supp1)
  PROC_K(2, supp2)
  PROC_K(3, supp3)
#undef PROC_K
  __syncthreads();

  // ---- rank kept boxes and emit outputs (fill everything: harness poisons d_out) ----
  float* out_s = out;
  float* out_c = out + BATCHN * MAX_DETC;
  float* out_b = out + 2 * BATCHN * MAX_DETC;
  const float*  clsb = scls + (size_t)b * NROW;
  const float4* boxb = (const float4*)sbox + (size_t)b * NROW;

  int base = 0;
  for (int w = 0; w < NW; ++w) {
    unsigned kwv = keptw[w];
    int i = w * 32 + lane;
    if ((kwv >> lane) & 1u) {
      int rank = base + __popc(kwv & ((1u << lane) - 1u));
      if (rank < MAX_DETC) {
        int oi = b * MAX_DETC + rank;
        out_s[oi] = keyb[i];
        out_c[oi] = clsb[i];
        float4 bb = boxb[i];
        float* ob = out_b + (size_t)oi * 4;
        ob[0] = bb.x; ob[1] = bb.y; ob[2] = bb.z; ob[3] = bb.w;
      }
    }
    base += __popc(kwv);
  }
  int total = base < MAX_DETC ? base : MAX_DETC;
  for (int r = total + lane; r < MAX_DETC; r += 32) {
    int oi = b * MAX_DETC + r;
    out_s[oi] = -1.f;
    out_c[oi] = -1.f;
    float* ob = out_b + (size_t)oi * 4;
    ob[0] = -1.f; ob[1] = -1.f; ob[2] = -1.f; ob[3] = -1.f;
  }
}

// ---------------- host launcher ----------------
extern "C" void kernel_launch(void* const* d_in, const int* in_sizes, int n_in,
                              void* d_out, int out_size, void* d_ws, size_t ws_size,
                              hipStream_t stream)
{
  (void)in_sizes; (void)n_in; (void)out_size; (void)ws_size;
  static const int HW[5]    = {128, 64, 32, 16, 8};
  static const int STR[5]   = {8, 16, 32, 64, 128};
  static const int BASE[5]  = {0, 16384, 20480, 21504, 21760};

  // workspace layout (~32.4 MB, all offsets 16B-aligned)
  float*    score  = (float*)d_ws;
  float*    kls    = score  + (size_t)BATCHN * NANCH;
  float*    box    = kls    + (size_t)BATCHN * NANCH;
  float*    cscore = box    + (size_t)BATCHN * NANCH * 4;
  unsigned* cpay   = (unsigned*)(cscore + (size_t)BATCHN * NCAND);
  float*    skey   = (float*)(cpay + (size_t)BATCHN * NCAND);
  float*    sboxp  = skey   + (size_t)BATCHN * NROW;
  float*    sclsp  = sboxp  + (size_t)BATCHN * NROW * 4;
  unsigned* maskp  = (unsigned*)(sclsp + (size_t)BATCHN * NROW);

  for (int l = 0; l < 5; l++) {
    const float* cls = (const float*)d_in[4 * l + 0];
    const float* reg = (const float*)d_in[4 * l + 1];
    const float* ctr = (const float*)d_in[4 * l + 2];
    // position input d_in[4*l+3] recomputed analytically in-kernel
    int total = BATCHN * HW[l] * HW[l];
    decode_kernel<<<(total + 255) / 256, 256, 0, stream>>>(
        cls, reg, ctr, HW[l], HW[l], STR[l], BASE[l], score, kls, box);
  }
  topk_kernel<<<BATCHN, 512, 0, stream>>>(score, 16384,     0,    0, cscore, cpay);
  topk_kernel<<<BATCHN, 512, 0, stream>>>(score,  4096, 16384, 1000, cscore, cpay);
  topk_kernel<<<BATCHN, 512, 0, stream>>>(score,  1024, 20480, 2000, cscore, cpay);
  copy_tail_kernel<<<(BATCHN * 320 + 255) / 256, 256, 0, stream>>>(score, cscore, cpay);
  sortgather_kernel<<<BATCHN, 512, 0, stream>>>(cscore, cpay, box, kls, skey, sboxp, sclsp);
  iou_mask_kernel<<<BATCHN * NW, 256, 0, stream>>>(sboxp, maskp);
  nms_out_kernel<<<BATCHN, 32, 0, stream>>>(maskp, skey, sboxp, sclsp, (float*)d_out);
}